// GAT_28724741275648
// MI455X (gfx1250) — compile-verified
//
#include <hip/hip_runtime.h>
#include <hip/hip_bf16.h>
#include <math.h>

#define N_NODES 100000
#define E_EDGES 3200000
#define F_IN    256
#define HID     8
#define HEADS   8
#define C1      64          // HEADS*HID
#define ETOT    (E_EDGES + N_NODES)

typedef float v2f __attribute__((ext_vector_type(2)));
typedef float v8f __attribute__((ext_vector_type(8)));

// ---------------------------------------------------------------------------
// Kernel 1: h1 = x @ W1   (fp32 WMMA 16x16x4, one wave per 16x64 output tile)
// W1 is staged into LDS in *fragment-major* order so each lane's B fragment
// (its two K-values) is a contiguous 8B ds_load_b64 -> no VGPR shuffle movs.
// LDS layout: frag = ((k/4)*4 + j)*32 + lane  holds
//   { W1[k+koff][j*16+l16], W1[k+koff+1][j*16+l16] },  koff=(lane>>4)*2
// ---------------------------------------------------------------------------
__global__ __launch_bounds__(256) void gat_gemm1_wmma(
    const float* __restrict__ x, const float* __restrict__ W1,
    float* __restrict__ h1)
{
    __shared__ float Wlds[F_IN * C1];                  // 64 KB, 8192 fragments
    const int t = threadIdx.x;
    #pragma unroll 4
    for (int i = 0; i < 32; ++i) {                     // 8192 frags / 256 thr
        int frag = t + i * 256;
        int l    = frag & 31;
        int j    = (frag >> 5) & 3;
        int ks   = frag >> 7;                          // k/4
        int koff = (l >> 4) * 2;
        int col  = j * 16 + (l & 15);
        int krow = ks * 4 + koff;
        float2 fr = make_float2(W1[krow * C1 + col], W1[(krow + 1) * C1 + col]);
        *(float2*)&Wlds[(size_t)frag * 2] = fr;
    }
    __syncthreads();

    const int wave   = t >> 5;                         // 8 waves / block
    const int lane   = t & 31;
    const int m_base = (blockIdx.x * 8 + wave) * 16;
    if (m_base >= N_NODES) return;

    const int half = lane >> 4;                        // 0 or 1
    const int l16  = lane & 15;
    const int koff = half * 2;                         // A frag K sub-offset
    const float* xrow = x + (size_t)(m_base + l16) * F_IN + koff;
    const float* wbase = &Wlds[lane * 2];

    v8f c0 = {}, c1 = {}, c2 = {}, c3 = {};
    for (int k = 0; k < F_IN; k += 4) {
        v2f a = *(const v2f*)(xrow + k);               // A: 16x4 fp32 frag
        const float* wb = wbase + (k >> 2) * 256;      // fragment-major LDS
        v2f b0 = *(const v2f*)(wb +   0);              // single ds_load_b64 each
        v2f b1 = *(const v2f*)(wb +  64);
        v2f b2 = *(const v2f*)(wb + 128);
        v2f b3 = *(const v2f*)(wb + 192);
        c0 = __builtin_amdgcn_wmma_f32_16x16x4_f32(false, a, false, b0, (short)0, c0, false, false);
        c1 = __builtin_amdgcn_wmma_f32_16x16x4_f32(false, a, false, b1, (short)0, c1, false, false);
        c2 = __builtin_amdgcn_wmma_f32_16x16x4_f32(false, a, false, b2, (short)0, c2, false, false);
        c3 = __builtin_amdgcn_wmma_f32_16x16x4_f32(false, a, false, b3, (short)0, c3, false, false);
    }
    // C/D layout: VGPR v, lanes 0-15 -> row m_base+v, lanes 16-31 -> row m_base+8+v
    #pragma unroll
    for (int v = 0; v < 8; ++v) {
        float* hr = h1 + (size_t)(m_base + half * 8 + v) * C1 + l16;
        hr[ 0] = c0[v];
        hr[16] = c1[v];
        hr[32] = c2[v];
        hr[48] = c3[v];
    }
}

// ---------------------------------------------------------------------------
// Kernel 2: per-(node,head) attention coefficients a_src, a_dst
// ---------------------------------------------------------------------------
__global__ void gat_attn1(const float* __restrict__ h1,
                          const float* __restrict__ att_src,
                          const float* __restrict__ att_dst,
                          float* __restrict__ as1, float* __restrict__ ad1)
{
    int tid = blockIdx.x * blockDim.x + threadIdx.x;
    if (tid >= N_NODES * HEADS) return;
    int n = tid >> 3, h = tid & 7;
    const float* hp = h1 + (size_t)n * C1 + h * HID;
    const float* sp = att_src + h * HID;
    const float* dp = att_dst + h * HID;
    float as = 0.f, ad = 0.f;
    #pragma unroll
    for (int c = 0; c < HID; ++c) { float v = hp[c]; as += v * sp[c]; ad += v * dp[c]; }
    as1[tid] = as; ad1[tid] = ad;
}

// ---------------------------------------------------------------------------
// Kernel 3: single edge pass, layer 1. thread = (edge, head).
//   w = exp(leaky_relu(a_s[src]+a_d[dst])); denom[dst] += w; acc[dst] += w*h1[src]
// (softmax max-shift skipped: |e| is O(1) here, exp cannot overflow; the
//  per-dst normalization divides out in gat_post1 -> one edge pass, not three)
// ---------------------------------------------------------------------------
__global__ void gat_edge1(const int* __restrict__ ei,
                          const float* __restrict__ as1, const float* __restrict__ ad1,
                          const float* __restrict__ h1,
                          float* __restrict__ denom1, float* __restrict__ acc1)
{
    int t = blockIdx.x * blockDim.x + threadIdx.x;        // < 26.4M, fits int
    if (t >= ETOT * HEADS) return;
    int e = t >> 3, h = t & 7;
    int s, d;
    if (e < E_EDGES) { s = ei[e]; d = ei[E_EDGES + e]; }
    else             { s = d = e - E_EDGES; }             // self-loop
    float a  = as1[s * 8 + h] + ad1[d * 8 + h];
    float el = a > 0.f ? a : 0.2f * a;                    // leaky_relu(0.2)
    float w  = __expf(el);
    atomicAdd(&denom1[d * 8 + h], w);
    const float* hp = h1 + (size_t)s * C1 + h * HID;      // coalesced across heads
    float* op       = acc1 + (size_t)d * C1 + h * HID;
    #pragma unroll
    for (int c = 0; c < HID; ++c) atomicAdd(&op[c], w * hp[c]);
}

// ---------------------------------------------------------------------------
// Kernel 4: normalize + bias + ELU (in place: acc1 -> h2)
// ---------------------------------------------------------------------------
__global__ void gat_post1(float* __restrict__ acc1, const float* __restrict__ denom1,
                          const float* __restrict__ b1)
{
    int tid = blockIdx.x * blockDim.x + threadIdx.x;
    if (tid >= N_NODES * C1) return;
    int n = tid >> 6, col = tid & 63, h = col >> 3;
    float v = acc1[tid] / (denom1[n * 8 + h] + 1e-16f) + b1[col];
    acc1[tid] = v > 0.f ? v : (__expf(v) - 1.0f);         // ELU
}

// ---------------------------------------------------------------------------
// Kernel 5: layer-2 GEMV z = h2 @ W2  (64 -> 1)
// ---------------------------------------------------------------------------
__global__ void gat_l2node(const float* __restrict__ h2, const float* __restrict__ W2,
                           float* __restrict__ z)
{
    int n = blockIdx.x * blockDim.x + threadIdx.x;
    if (n >= N_NODES) return;
    const float* hp = h2 + (size_t)n * C1;
    float acc = 0.f;
    #pragma unroll
    for (int j = 0; j < C1; ++j) acc += hp[j] * W2[j];
    z[n] = acc;
}

// ---------------------------------------------------------------------------
// Kernel 6: single edge pass, layer 2 (scalar head)
// ---------------------------------------------------------------------------
__global__ void gat_edge2(const int* __restrict__ ei, const float* __restrict__ z,
                          const float* __restrict__ asc, const float* __restrict__ adc,
                          float* __restrict__ denom2, float* __restrict__ num2)
{
    int e = blockIdx.x * blockDim.x + threadIdx.x;
    if (e >= ETOT) return;
    int s, d;
    if (e < E_EDGES) { s = ei[e]; d = ei[E_EDGES + e]; }
    else             { s = d = e - E_EDGES; }
    float zs = z[s];
    float a  = zs * asc[0] + z[d] * adc[0];
    float el = a > 0.f ? a : 0.2f * a;
    float w  = __expf(el);
    atomicAdd(&denom2[d], w);
    atomicAdd(&num2[d], w * zs);
}

// ---------------------------------------------------------------------------
// Kernel 7: out = sigmoid(num/denom + b2)
// ---------------------------------------------------------------------------
__global__ void gat_final(const float* __restrict__ num2, const float* __restrict__ denom2,
                          const float* __restrict__ b2, float* __restrict__ out)
{
    int n = blockIdx.x * blockDim.x + threadIdx.x;
    if (n >= N_NODES) return;
    float v = num2[n] / (denom2[n] + 1e-16f) + b2[0];
    out[n] = 1.0f / (1.0f + __expf(-v));
}

__global__ void gat_zero(float* __restrict__ p, int n)
{
    int i = blockIdx.x * blockDim.x + threadIdx.x;
    if (i < n) p[i] = 0.f;
}

// ---------------------------------------------------------------------------
extern "C" void kernel_launch(void* const* d_in, const int* in_sizes, int n_in,
                              void* d_out, int out_size, void* d_ws, size_t ws_size,
                              hipStream_t stream)
{
    const float* x    = (const float*)d_in[0];
    const int*   ei   = (const int*)  d_in[1];
    const float* W1   = (const float*)d_in[2];
    const float* as_w = (const float*)d_in[3];
    const float* ad_w = (const float*)d_in[4];
    const float* b1   = (const float*)d_in[5];
    const float* W2   = (const float*)d_in[6];
    const float* asc2 = (const float*)d_in[7];
    const float* adc2 = (const float*)d_in[8];
    const float* b2   = (const float*)d_in[9];
    float* out = (float*)d_out;

    float* ws = (float*)d_ws;
    // workspace layout (floats); zeroed regions are contiguous at the end
    float* h1     = ws;                           // N*64
    float* as1    = h1  + (size_t)N_NODES * C1;   // N*8
    float* ad1    = as1 + (size_t)N_NODES * HEADS;// N*8
    float* z      = ad1 + (size_t)N_NODES * HEADS;// N
    float* denom1 = z   + (size_t)N_NODES;        // N*8   (zeroed)
    float* acc1   = denom1 + (size_t)N_NODES * HEADS; // N*64 (zeroed), becomes h2
    float* denom2 = acc1 + (size_t)N_NODES * C1;  // N     (zeroed)
    float* num2   = denom2 + (size_t)N_NODES;     // N     (zeroed)
    const int n_zero = N_NODES * HEADS + N_NODES * C1 + 2 * N_NODES; // 7.4M

    const int B = 256;
    gat_zero <<<(n_zero + B - 1) / B, B, 0, stream>>>(denom1, n_zero);

    gat_gemm1_wmma<<<(N_NODES / 16 + 7) / 8, B, 0, stream>>>(x, W1, h1);

    gat_attn1<<<(N_NODES * HEADS + B - 1) / B, B, 0, stream>>>(h1, as_w, ad_w, as1, ad1);

    gat_edge1<<<(ETOT * HEADS + B - 1) / B, B, 0, stream>>>(ei, as1, ad1, h1, denom1, acc1);

    gat_post1<<<(N_NODES * C1 + B - 1) / B, B, 0, stream>>>(acc1, denom1, b1);

    gat_l2node<<<(N_NODES + B - 1) / B, B, 0, stream>>>(acc1, W2, z);

    gat_edge2<<<(ETOT + B - 1) / B, B, 0, stream>>>(ei, z, asc2, adc2, denom2, num2);

    gat_final<<<(N_NODES + B - 1) / B, B, 0, stream>>>(num2, denom2, b2, out);

    (void)in_sizes; (void)n_in; (void)out_size; (void)ws_size;
}